// GraphSAGE_82386062672069
// MI455X (gfx1250) — compile-verified
//
#include <hip/hip_runtime.h>

typedef __attribute__((ext_vector_type(2))) float v2f;
typedef __attribute__((ext_vector_type(8))) float v8f;
typedef __attribute__((ext_vector_type(4))) float f4;

#define N0C 10
#define N1C 25
#define DC 64
#define H1C 128
#define H0C 128
#define BTOT 4096

// ---------------- Kernel 1: gather + mean-agg + fc1 GEMM + n0-reduction ----
// block = 256 threads (8 waves), owns BPB=4 batch elems -> 40 real rows,
// padded to 48 rows (3 WMMA row tiles of 16).
#define BPB 4
#define RROWS 40          // real rows = BPB * N0
#define PROWS 48          // padded rows (3 x 16)
#define XS 132            // LDS row stride (pad: 132 % 64 = 4 -> conflict-free A reads)
#define HS 132

__global__ __launch_bounds__(256) void sage_k1(
    const int* __restrict__ neigh0,     // [B, N0]
    const int* __restrict__ neigh1,     // [B, N0, N1]
    const float* __restrict__ emb,      // [NODE_CNT+1, 64]
    const float* __restrict__ W1,       // [128, 128]
    float* __restrict__ agg0)           // [B, 128] (workspace)
{
    __shared__ float Xl[PROWS * XS];    // [e_n | agg1] rows
    __shared__ float Hl[PROWS * HS];    // h1 rows

    const int t  = threadIdx.x;
    const int wg = blockIdx.x;

    // ---- Phase A: gather. half-wave (16 lanes) per row, lane covers 4 dims.
    const int half = t >> 4;            // 0..15
    const int hl   = t & 15;            // lane within half
    for (int q = 0; q < 3; ++q) {
        const int r = q * 16 + half;    // 0..47
        if (r < RROWS) {
            const int grow = wg * RROWS + r;        // global (b, n0) row
            // e_n = embed_table[neigh0[grow]]
            const int idx0 = neigh0[grow];
            f4 e = ((const f4*)(emb + (long)idx0 * DC))[hl];
            *(f4*)&Xl[r * XS + hl * 4] = e;
            // agg1 = mean over 25 neighbor embeddings
            f4 acc; acc.x = 0.f; acc.y = 0.f; acc.z = 0.f; acc.w = 0.f;
            const int nbase = grow * N1C;
            for (int j = 0; j < N1C; ++j) {
                const int idx = neigh1[nbase + j];
                f4 v = ((const f4*)(emb + (long)idx * DC))[hl];
                acc.x += v.x; acc.y += v.y; acc.z += v.z; acc.w += v.w;
            }
            const float s = 1.0f / 25.0f;
            acc.x *= s; acc.y *= s; acc.z *= s; acc.w *= s;
            *(f4*)&Xl[r * XS + 64 + hl * 4] = acc;
        } else {
            // dummy rows: zero so WMMA consumes clean data
            f4 z; z.x = 0.f; z.y = 0.f; z.z = 0.f; z.w = 0.f;
            *(f4*)&Xl[r * XS + hl * 4] = z;
            *(f4*)&Xl[r * XS + 64 + hl * 4] = z;
        }
    }
    __syncthreads();

    // ---- Phase B: H[48x128] = X[48x128] @ W1[128x128] via f32 WMMA 16x16x4.
    const int wv   = t >> 5;            // wave id 0..7 -> column block
    const int lane = t & 31;
    const int l16  = lane & 15;
    const int hi   = lane >> 4;         // half-wave select (K offset +2)
    const int n    = wv * 16 + l16;     // output column
    for (int rt = 0; rt < 3; ++rt) {
        const float* arow = &Xl[(rt * 16 + l16) * XS];   // A row M = l16
        v8f acc = {};
        for (int kb = 0; kb < 128; kb += 4) {
            const int ka = kb + 2 * hi;
            v2f a; a.x = arow[ka];              a.y = arow[ka + 1];
            v2f b; b.x = W1[ka * H1C + n];      b.y = W1[(ka + 1) * H1C + n];
            acc = __builtin_amdgcn_wmma_f32_16x16x4_f32(
                false, a, false, b, (short)0, acc, false, false);
        }
        // C/D layout: VGPR j -> row M = j + 8*hi, col N = n
        for (int j = 0; j < 8; ++j)
            Hl[(rt * 16 + j + 8 * hi) * HS + n] = acc[j];
    }
    __syncthreads();

    // ---- Phase C: agg0[b] = mean over n0 of h1 rows (deterministic).
    for (int i = 0; i < 2; ++i) {
        const int o  = t + i * 256;     // 0..511 = BPB*128 outputs
        const int bl = o >> 7;          // local b
        const int h  = o & 127;
        float s = 0.f;
        for (int j = 0; j < N0C; ++j) s += Hl[(bl * N0C + j) * HS + h];
        agg0[(long)(wg * BPB + bl) * H1C + h] = s * (1.0f / 10.0f);
    }
}

// ---------------- Kernel 2: fc0 = sigmoid([e_v | agg0] @ W0 + b0) ----------
#define YS 196              // 192 + pad (196 % 64 = 4)

__global__ __launch_bounds__(256) void sage_k2(
    const int* __restrict__ inputs,     // [B]
    const float* __restrict__ emb,      // [NODE_CNT+1, 64]
    const float* __restrict__ agg0,     // [B, 128] (workspace)
    const float* __restrict__ W0,       // [192, 128]
    const float* __restrict__ b0,       // [128]
    float* __restrict__ out)            // [B, 128]
{
    __shared__ float Yl[16 * YS];

    const int t  = threadIdx.x;
    const int bb = blockIdx.x * 16;

    // stage e_v rows: 16 half-waves, one row each
    const int half = t >> 4;
    const int hl   = t & 15;
    {
        const int idx = inputs[bb + half];
        f4 e = ((const f4*)(emb + (long)idx * DC))[hl];
        *(f4*)&Yl[half * YS + hl * 4] = e;
    }
    // stage agg0 rows: 16*128 floats = 512 float4
    for (int i = 0; i < 2; ++i) {
        const int g   = t + i * 256;
        const int row = g >> 5;
        const int c4  = (g & 31) << 2;
        f4 v = ((const f4*)(agg0 + (long)bb * H1C))[g];
        *(f4*)&Yl[row * YS + 64 + c4] = v;
    }
    __syncthreads();

    const int wv   = t >> 5;
    const int lane = t & 31;
    const int l16  = lane & 15;
    const int hi   = lane >> 4;
    const int n    = wv * 16 + l16;
    const float* arow = &Yl[l16 * YS];
    v8f acc = {};
    for (int kb = 0; kb < 192; kb += 4) {
        const int ka = kb + 2 * hi;
        v2f a; a.x = arow[ka];              a.y = arow[ka + 1];
        v2f b; b.x = W0[ka * H0C + n];      b.y = W0[(ka + 1) * H0C + n];
        acc = __builtin_amdgcn_wmma_f32_16x16x4_f32(
            false, a, false, b, (short)0, acc, false, false);
    }
    const float bias = b0[n];
    for (int j = 0; j < 8; ++j) {
        const float x = acc[j] + bias;
        const float y = 1.0f / (1.0f + __expf(-x));
        out[(long)(bb + j + 8 * hi) * H0C + n] = y;
    }
}

extern "C" void kernel_launch(void* const* d_in, const int* in_sizes, int n_in,
                              void* d_out, int out_size, void* d_ws, size_t ws_size,
                              hipStream_t stream) {
    (void)in_sizes; (void)n_in; (void)out_size; (void)ws_size;
    const int*   inputs = (const int*)d_in[0];     // [4096]
    const int*   neigh0 = (const int*)d_in[1];     // [4096, 10]
    const int*   neigh1 = (const int*)d_in[2];     // [4096, 10, 25]
    const float* emb    = (const float*)d_in[3];   // [1000001, 64]
    const float* W1     = (const float*)d_in[4];   // [128, 128]
    const float* W0     = (const float*)d_in[5];   // [192, 128]
    const float* b0     = (const float*)d_in[6];   // [128]
    float* out  = (float*)d_out;                   // [4096, 128]
    float* agg0 = (float*)d_ws;                    // [4096, 128] scratch (fully overwritten)

    sage_k1<<<BTOT / BPB, 256, 0, stream>>>(neigh0, neigh1, emb, W1, agg0);
    sage_k2<<<BTOT / 16,  256, 0, stream>>>(inputs, emb, agg0, W0, b0, out);
}